// GraphEncoder_29618094473824
// MI455X (gfx1250) — compile-verified
//
#include <hip/hip_runtime.h>
#include <hip/hip_bf16.h>

typedef __attribute__((ext_vector_type(2))) float v2f;
typedef __attribute__((ext_vector_type(8))) float v8f;

// ---------------------------------------------------------------- degree prep
__global__ void k_init_deg(float* __restrict__ deg, int N) {
  int i = blockIdx.x * blockDim.x + threadIdx.x;
  if (i < N) deg[i] = 1.0f;                       // self-loop
}

__global__ void k_count_deg(const int* __restrict__ dst, float* __restrict__ deg, int E) {
  int i = blockIdx.x * blockDim.x + threadIdx.x;
  if (i < E) unsafeAtomicAdd(&deg[dst[i]], 1.0f);
}

__global__ void k_deg_to_dinv(float* __restrict__ deg, int N) {
  int i = blockIdx.x * blockDim.x + threadIdx.x;
  if (i < N) { float d = deg[i]; deg[i] = d > 0.0f ? rsqrtf(d) : 0.0f; }
}

// ---------------------------------------------------------------- WMMA GEMM
// T[N,128] = (RELU_IN ? relu(X) : X)[N,128] @ W[128,128] + bias[128]
// 256 threads = 8 waves; wave w computes rows [blk*128 + 16w, +16) x all 128 cols
// using V_WMMA_F32_16X16X4_F32 (exact fp32 math, K stepped by 4).
template <bool RELU_IN>
__global__ __launch_bounds__(256) void k_gemm_bias(const float* __restrict__ X,
                                                   const float* __restrict__ W,
                                                   const float* __restrict__ bias,
                                                   float* __restrict__ T, int N) {
  __shared__ float sW[32][128];                   // 16 KB K-slab of W
  const int lane = threadIdx.x & 31;
  const int wv   = threadIdx.x >> 5;              // 0..7
  const int hl   = lane >> 4;                     // half-wave: 0/1
  const int m    = lane & 15;
  const int row0 = blockIdx.x * 128 + wv * 16;
  const int arow = row0 + m;                      // row this lane loads A for
  const bool aok = arow < N;
  const size_t abase = (size_t)(aok ? arow : 0) * 128;

  const v8f vzero = {0.f, 0.f, 0.f, 0.f, 0.f, 0.f, 0.f, 0.f};
  v8f acc[8];
#pragma unroll
  for (int t = 0; t < 8; ++t) acc[t] = vzero;

  for (int kb = 0; kb < 128; kb += 32) {
    __syncthreads();
    {   // cooperative slab load: 32x128 floats = 1024 float4, 4 per thread
      const float4* s4 = (const float4*)(W + kb * 128);
      float4* d4 = (float4*)(&sW[0][0]);
#pragma unroll
      for (int i = 0; i < 4; ++i) d4[threadIdx.x + i * 256] = s4[threadIdx.x + i * 256];
    }
    __syncthreads();
#pragma unroll
    for (int kl = 0; kl < 32; kl += 4) {
      const int kk = kl + 2 * hl;                 // lane's K pair within slab
      // A fragment: lane(M=m, half hl) holds X[arow][kb+kk], X[arow][kb+kk+1]
      const float2 av = *(const float2*)(X + abase + (kb + kk));
      float ax = aok ? av.x : 0.0f;
      float ay = aok ? av.y : 0.0f;
      if (RELU_IN) { ax = fmaxf(ax, 0.0f); ay = fmaxf(ay, 0.0f); }
      v2f a; a.x = ax; a.y = ay;
#pragma unroll
      for (int t = 0; t < 8; ++t) {
        // B fragment: VGPR v holds W[kb+kk+v][t*16+m]  (K = v + 2*(lane/16))
        v2f b;
        b.x = sW[kk][t * 16 + m];
        b.y = sW[kk + 1][t * 16 + m];
        acc[t] = __builtin_amdgcn_wmma_f32_16x16x4_f32(
            /*neg_a=*/false, a, /*neg_b=*/false, b,
            /*c_mod=*/(short)0, acc[t], /*reuse_a=*/false, /*reuse_b=*/false);
      }
    }
  }
  // C/D layout: VGPR v, lane -> row = row0 + v + 8*hl, col = t*16 + m
  const int rbase = row0 + 8 * hl;
#pragma unroll
  for (int t = 0; t < 8; ++t) {
    const int col = t * 16 + m;
    const float bv = bias[col];
#pragma unroll
    for (int v = 0; v < 8; ++v) {
      const int r = rbase + v;
      if (r < N) T[(size_t)r * 128 + col] = acc[t][v] + bv;
    }
  }
}

// ---------------------------------------------------------------- aggregation
// A[i][:] = dinv[i]^2 * T[i][:]   (self-loop term; also zero-initializes A)
__global__ void k_selfloop(const float* __restrict__ T, const float* __restrict__ dinv,
                           float* __restrict__ A, int N) {
  int idx = blockIdx.x * blockDim.x + threadIdx.x;
  if (idx >= N * 32) return;
  int i = idx >> 5;
  int f = (idx & 31) << 2;
  float s = dinv[i] * dinv[i];
  float4 v = *(const float4*)(T + (size_t)i * 128 + f);
  float4 o; o.x = s * v.x; o.y = s * v.y; o.z = s * v.z; o.w = s * v.w;
  *(float4*)(A + (size_t)i * 128 + f) = o;
}

// A[dst][:] += dinv[src]*dinv[dst] * T[src][:]   (float4 per thread, L2-resident)
__global__ void k_edge_agg(const int* __restrict__ ei, const float* __restrict__ dinv,
                           const float* __restrict__ T, float* __restrict__ A, int E) {
  int idx = blockIdx.x * blockDim.x + threadIdx.x;
  if (idx >= E * 32) return;                       // E*32 = 16M < 2^31
  int e = idx >> 5;
  int f = (idx & 31) << 2;
  int s = ei[e];
  int d = ei[E + e];
  float nrm = dinv[s] * dinv[d];
  float4 v = *(const float4*)(T + (size_t)s * 128 + f);
  float* out = A + (size_t)d * 128 + f;
  unsafeAtomicAdd(out + 0, nrm * v.x);
  unsafeAtomicAdd(out + 1, nrm * v.y);
  unsafeAtomicAdd(out + 2, nrm * v.z);
  unsafeAtomicAdd(out + 3, nrm * v.w);
}

// ---------------------------------------------------------------- layer norm
// one wave32 per row of 128; in-place on io
__global__ __launch_bounds__(256) void k_layernorm(float* __restrict__ io,
                                                   const float* __restrict__ gamma,
                                                   const float* __restrict__ beta, int N) {
  const int lane = threadIdx.x & 31;
  const int wv   = threadIdx.x >> 5;
  const int row  = blockIdx.x * 8 + wv;
  if (row >= N) return;                            // uniform per wave
  const int f = lane << 2;
  float4 v = *(const float4*)(io + (size_t)row * 128 + f);
  float s = v.x + v.y + v.z + v.w;
#pragma unroll
  for (int msk = 16; msk >= 1; msk >>= 1) s += __shfl_xor(s, msk, 32);
  const float mu = s * (1.0f / 128.0f);
  float dx = v.x - mu, dy = v.y - mu, dz = v.z - mu, dw = v.w - mu;
  float q = dx * dx + dy * dy + dz * dz + dw * dw;
#pragma unroll
  for (int msk = 16; msk >= 1; msk >>= 1) q += __shfl_xor(q, msk, 32);
  const float inv = rsqrtf(q * (1.0f / 128.0f) + 1e-5f);
  float4 g = *(const float4*)(gamma + f);
  float4 b = *(const float4*)(beta + f);
  float4 o;
  o.x = dx * inv * g.x + b.x;
  o.y = dy * inv * g.y + b.y;
  o.z = dz * inv * g.z + b.z;
  o.w = dw * inv * g.w + b.w;
  *(float4*)(io + (size_t)row * 128 + f) = o;
}

// ---------------------------------------------------------------- launch
extern "C" void kernel_launch(void* const* d_in, const int* in_sizes, int n_in,
                              void* d_out, int out_size, void* d_ws, size_t ws_size,
                              hipStream_t stream) {
  const float* x  = (const float*)d_in[0];
  const int*   ei = (const int*)d_in[1];   // [2,E] int32 (JAX x64 disabled)
  const float* W1 = (const float*)d_in[2];
  const float* b1 = (const float*)d_in[3];
  const float* W2 = (const float*)d_in[4];
  const float* b2 = (const float*)d_in[5];
  const float* gm = (const float*)d_in[6];
  const float* bt = (const float*)d_in[7];
  float* out = (float*)d_out;

  const int N = in_sizes[0] / 128;
  const int E = in_sizes[1] / 2;

  // workspace: dinv[N] | T[N*128] | A[N*128]   (~51.4 MB)
  float* dinv = (float*)d_ws;
  size_t off = (((size_t)N * sizeof(float)) + 255) & ~(size_t)255;
  float* T = (float*)((char*)d_ws + off);
  float* A = T + (size_t)N * 128;

  dim3 blk(256);
  int gN  = (N + 255) / 256;
  int gE  = (E + 255) / 256;
  int gNF = (int)(((size_t)N * 32 + 255) / 256);
  int gEF = (int)(((size_t)E * 32 + 255) / 256);
  int gG  = (N + 127) / 128;
  int gLN = (N + 7) / 8;

  // degree -> dinv
  k_init_deg<<<gN, blk, 0, stream>>>(dinv, N);
  k_count_deg<<<gE, blk, 0, stream>>>(ei + E, dinv, E);
  k_deg_to_dinv<<<gN, blk, 0, stream>>>(dinv, N);

  // layer 1: T = x@W1+b1 ; A = aggregate(T)
  k_gemm_bias<false><<<gG, blk, 0, stream>>>(x, W1, b1, T, N);
  k_selfloop<<<gNF, blk, 0, stream>>>(T, dinv, A, N);
  k_edge_agg<<<gEF, blk, 0, stream>>>(ei, dinv, T, A, E);

  // layer 2: T = relu(A)@W2+b2 ; out = aggregate(T)
  k_gemm_bias<true><<<gG, blk, 0, stream>>>(A, W2, b2, T, N);
  k_selfloop<<<gNF, blk, 0, stream>>>(T, dinv, out, N);
  k_edge_agg<<<gEF, blk, 0, stream>>>(ei, dinv, T, out, E);

  // layer norm in place on out
  k_layernorm<<<gLN, blk, 0, stream>>>(out, gm, bt, N);
}